// GINGAT_89232240542332
// MI455X (gfx1250) — compile-verified
//
#include <hip/hip_runtime.h>
#include <hip/hip_bf16.h>

typedef __attribute__((ext_vector_type(2))) float v2f;
typedef __attribute__((ext_vector_type(8))) float v8f;

static constexpr int NN   = 32768;    // nodes
static constexpr int NE   = 262144;   // edges
static constexpr int BG   = 256;      // graphs
static constexpr int TT   = 128;      // nodes per graph (GRU steps)
static constexpr int N7   = BG * 7;   // GAT star-graph nodes

// ---------------- utility ----------------
__global__ void k_zero(float* __restrict__ p, long n) {
  long i = (long)blockIdx.x * blockDim.x + threadIdx.x;
  if (i < n) p[i] = 0.0f;
}

// ---------------- GINE edge message + scatter-add ----------------
// agg[dst] += relu(h[src] + We@ea + be); one thread = (edge, 8-channel chunk)
__global__ void k_gine_scatter(const float* __restrict__ h,
                               const int* __restrict__ src,
                               const int* __restrict__ dst,
                               const float* __restrict__ ea,   // [E,16]
                               const float* __restrict__ We,   // [D,16]
                               const float* __restrict__ be,   // [D]
                               float* __restrict__ agg,        // [N,D]
                               int E, int D) {
  int t = blockIdx.x * blockDim.x + threadIdx.x;
  int chunks = D >> 3;
  int e = t / chunks;
  if (e >= E) return;
  int c0 = (t - e * chunks) << 3;
  int s = src[e], d = dst[e];
  float eav[16];
#pragma unroll
  for (int k = 0; k < 16; ++k) eav[k] = ea[(size_t)e * 16 + k];
  const float* hrow = h + (size_t)s * D;
  float* arow = agg + (size_t)d * D;
#pragma unroll
  for (int j = 0; j < 8; ++j) {
    int c = c0 + j;
    float lin = be[c];
    const float* wr = We + (size_t)c * 16;
#pragma unroll
    for (int k = 0; k < 16; ++k) lin = fmaf(wr[k], eav[k], lin);
    float m = fmaxf(hrow[c] + lin, 0.0f);
    atomicAdd(&arow[c], m);
  }
}

// ---------------- fp32 WMMA GEMM, 16x64 tile per wave ----------------
// C[M,N] = act((A (+A2))[M,K(lda)] @ W[N,K]^T + bias).
// One wave computes a 16x64 output block via 4 independent accumulators:
// A fragment loaded once per k-step, reused across 4 B fragments -> 4
// independent v_wmma chains for XDL co-execution, 4x less A traffic.
// HAS_A2 is a compile-time flag so the inner loop is branch-free.
// fp32 WMMA layouts (ISA 7.12.2):
//   A 16x4 : element (m,k) -> lane = m + 16*(k>>1), vgpr = k&1
//   B 4x16 : element (k,n) -> lane = n + 16*(k>>1), vgpr = k&1
//   C 16x16: vgpr j -> (M=j, N=lane) lanes 0-15, (M=j+8, N=lane-16) lanes 16-31
// Requires: M % 16 == 0, N % 64 == 0, K % 4 == 0.
template <bool HAS_A2>
__global__ __launch_bounds__(256)
void k_wmma_gemm(const float* __restrict__ A,
                 const float* __restrict__ A2,
                 const float* __restrict__ W,
                 const float* __restrict__ bias,
                 float* __restrict__ C,
                 int M, int N, int K, int lda, int act) {
  int wave = (blockIdx.x * blockDim.x + threadIdx.x) >> 5;
  int lane = threadIdx.x & 31;
  int nblk = N >> 6;          // 64-wide N blocks
  int mtiles = M >> 4;
  if (wave >= mtiles * nblk) return;
  int mt = wave / nblk;
  int nb = wave - mt * nblk;
  int half = lane >> 4;       // which K-pair this lane holds
  int l15  = lane & 15;
  const float* arow  = A + (size_t)(mt * 16 + l15) * lda;
  const float* a2row = HAS_A2 ? A2 + (size_t)(mt * 16 + l15) * lda : nullptr;
  const float* w0 = W + (size_t)(nb * 64 +  0 + l15) * K;
  const float* w1 = W + (size_t)(nb * 64 + 16 + l15) * K;
  const float* w2 = W + (size_t)(nb * 64 + 32 + l15) * K;
  const float* w3 = W + (size_t)(nb * 64 + 48 + l15) * K;
  v8f acc0 = {}, acc1 = {}, acc2 = {}, acc3 = {};
  int kofs = half * 2;
#pragma unroll 2
  for (int kk = 0; kk < K; kk += 4) {
    int ka = kk + kofs;
    v2f a;
    a.x = arow[ka];
    a.y = arow[ka + 1];
    if (HAS_A2) { a.x += a2row[ka]; a.y += a2row[ka + 1]; }
    v2f b0; b0.x = w0[ka]; b0.y = w0[ka + 1];
    v2f b1; b1.x = w1[ka]; b1.y = w1[ka + 1];
    v2f b2; b2.x = w2[ka]; b2.y = w2[ka + 1];
    v2f b3; b3.x = w3[ka]; b3.y = w3[ka + 1];
    acc0 = __builtin_amdgcn_wmma_f32_16x16x4_f32(false, a, false, b0, (short)0, acc0, false, false);
    acc1 = __builtin_amdgcn_wmma_f32_16x16x4_f32(false, a, false, b1, (short)0, acc1, false, false);
    acc2 = __builtin_amdgcn_wmma_f32_16x16x4_f32(false, a, false, b2, (short)0, acc2, false, false);
    acc3 = __builtin_amdgcn_wmma_f32_16x16x4_f32(false, a, false, b3, (short)0, acc3, false, false);
  }
  float* crow = C + (size_t)(mt * 16 + half * 8) * N + nb * 64 + l15;
  v8f accs[4] = { acc0, acc1, acc2, acc3 };
#pragma unroll
  for (int t = 0; t < 4; ++t) {
    int col = nb * 64 + t * 16 + l15;
    float bv = bias ? bias[col] : 0.0f;
#pragma unroll
    for (int j = 0; j < 8; ++j) {
      float v = accs[t][j] + bv;
      if (act == 1) v = fmaxf(v, 0.0f);
      crow[(size_t)j * N + t * 16] = v;
    }
  }
}

// ---------------- training-mode BatchNorm over rows ----------------
__global__ void k_bn_stats(const float* __restrict__ X, int M, int N,
                           const float* __restrict__ g, const float* __restrict__ b,
                           float* __restrict__ scale, float* __restrict__ shift) {
  int c = blockIdx.x;            // one block per channel
  __shared__ float s1[256], s2[256];
  float sum = 0.f, sq = 0.f;
  for (int r = threadIdx.x; r < M; r += blockDim.x) {
    float v = X[(size_t)r * N + c];
    sum += v; sq += v * v;
  }
  s1[threadIdx.x] = sum; s2[threadIdx.x] = sq;
  __syncthreads();
  for (int st = 128; st > 0; st >>= 1) {
    if ((int)threadIdx.x < st) { s1[threadIdx.x] += s1[threadIdx.x + st];
                                 s2[threadIdx.x] += s2[threadIdx.x + st]; }
    __syncthreads();
  }
  if (threadIdx.x == 0) {
    float m = s1[0] / (float)M;
    float v = s2[0] / (float)M - m * m;
    float inv = rsqrtf(v + 1e-5f);
    float sc = g[c] * inv;
    scale[c] = sc;
    shift[c] = b[c] - m * sc;
  }
}

__global__ void k_bn_relu(float* __restrict__ X, long n, int N,
                          const float* __restrict__ scale,
                          const float* __restrict__ shift) {
  long i = (long)blockIdx.x * blockDim.x + threadIdx.x;
  if (i >= n) return;
  int c = (int)(i & (N - 1));   // N is a power of two (256)
  X[i] = fmaxf(X[i] * scale[c] + shift[c], 0.0f);
}

// ---------------- GRU gates ----------------
__global__ void k_gru_gate(const float* __restrict__ GI,   // [NN,768], row = b*128+t
                           const float* __restrict__ GH,   // [BG,768]
                           float* __restrict__ hstate,     // [BG,256]
                           float* __restrict__ hs,         // [BG,128,256]
                           int t) {
  int i = blockIdx.x * blockDim.x + threadIdx.x;   // 0..65535
  int b = i >> 8, d = i & 255;
  const float* gi = GI + (size_t)(b * TT + t) * 768;
  const float* gh = GH + (size_t)b * 768;
  float r = 1.0f / (1.0f + expf(-(gi[d]       + gh[d])));
  float z = 1.0f / (1.0f + expf(-(gi[d + 256] + gh[d + 256])));
  float n = tanhf(gi[d + 512] + r * gh[d + 512]);
  float hprev = hstate[b * 256 + d];
  float hnew = (1.0f - z) * n + z * hprev;
  hstate[b * 256 + d] = hnew;
  hs[((size_t)b * TT + t) * 256 + d] = hnew;
}

// ---------------- attention pooling over time ----------------
__global__ void k_attn_pool(const float* __restrict__ hs,   // [BG,128,256]
                            const float* __restrict__ attw, // [256]
                            const float* __restrict__ attb, // [1]
                            float* __restrict__ g) {        // [BG,256]
  int b = blockIdx.x, tid = threadIdx.x;
  __shared__ float a[128];
  __shared__ float red[128];
  if (tid < 128) {
    const float* row = hs + ((size_t)b * TT + tid) * 256;
    float s = attb[0];
    for (int k = 0; k < 256; ++k) s += row[k] * attw[k];
    a[tid] = s;
  }
  __syncthreads();
  if (tid < 128) red[tid] = a[tid];
  __syncthreads();
  for (int st = 64; st > 0; st >>= 1) {
    if (tid < st) red[tid] = fmaxf(red[tid], red[tid + st]);
    __syncthreads();
  }
  float mx = red[0];
  __syncthreads();
  if (tid < 128) { a[tid] = expf(a[tid] - mx); red[tid] = a[tid]; }
  __syncthreads();
  for (int st = 64; st > 0; st >>= 1) {
    if (tid < st) red[tid] += red[tid + st];
    __syncthreads();
  }
  float den = red[0];
  const float* base = hs + (size_t)b * TT * 256;
  float s = 0.f;
  for (int t = 0; t < TT; ++t) s += a[t] * base[(size_t)t * 256 + tid];
  g[b * 256 + tid] = s / den;
}

// ---------------- assemble GAT input rows ----------------
__global__ void k_build_xd(const float* __restrict__ g,      // [BG,256]
                           const float* __restrict__ feats,  // [6,BG,256]
                           float* __restrict__ xd) {         // [N7,256]
  int i = blockIdx.x * blockDim.x + threadIdx.x;
  if (i >= N7 * 256) return;
  int d = i & 255, rest = i >> 8;
  int j = rest % 7, b = rest / 7;
  xd[i] = (j == 0) ? g[b * 256 + d]
                   : feats[((size_t)(j - 1) * BG + b) * 256 + d];
}

// ---------------- GATv2 per-node combine (star graphs, <=2 in-edges) ----------
__global__ void k_gat_node(const float* __restrict__ xl,   // [N7,1024]
                           const float* __restrict__ xr,   // [N7,1024]
                           const float* __restrict__ att,  // [4,256]
                           const float* __restrict__ gbias,// [256]
                           float* __restrict__ out) {      // [N7,256]
  int i = blockIdx.x;
  int b = i / 7, j = i - b * 7;
  int csrc = b * 7;
  int c = threadIdx.x;
  bool hasStar = (j != 0);
  __shared__ float red[256];
  __shared__ float A1[4], A2h[4];
  for (int h = 0; h < 4; ++h) {
    float e2 = xl[(size_t)i * 1024 + h * 256 + c] + xr[(size_t)i * 1024 + h * 256 + c];
    e2 = (e2 > 0.f) ? e2 : 0.2f * e2;
    float p2 = att[h * 256 + c] * e2;
    float p1 = 0.f;
    if (hasStar) {
      float e1 = xl[(size_t)csrc * 1024 + h * 256 + c] + xr[(size_t)i * 1024 + h * 256 + c];
      e1 = (e1 > 0.f) ? e1 : 0.2f * e1;
      p1 = att[h * 256 + c] * e1;
    }
    red[c] = p2; __syncthreads();
    for (int st = 128; st > 0; st >>= 1) { if (c < st) red[c] += red[c + st]; __syncthreads(); }
    if (c == 0) A2h[h] = red[0];
    __syncthreads();
    red[c] = p1; __syncthreads();
    for (int st = 128; st > 0; st >>= 1) { if (c < st) red[c] += red[c + st]; __syncthreads(); }
    if (c == 0) A1[h] = red[0];
    __syncthreads();
  }
  if (c < 4) {
    if (hasStar) {
      float L1 = A1[c], L2 = A2h[c];
      float mx = fmaxf(L1, L2);
      float a1 = expf(L1 - mx), a2 = expf(L2 - mx);
      float den = a1 + a2;
      A1[c] = a1 / den; A2h[c] = a2 / den;
    } else { A1[c] = 0.f; A2h[c] = 1.f; }
  }
  __syncthreads();
  float s = 0.f;
#pragma unroll
  for (int h = 0; h < 4; ++h) {
    float v = A2h[h] * xl[(size_t)i * 1024 + h * 256 + c];
    if (hasStar) v += A1[h] * xl[(size_t)csrc * 1024 + h * 256 + c];
    s += v;
  }
  out[(size_t)i * 256 + c] = s * 0.25f + gbias[c];
}

// ---------------- final tiny FC2 ----------------
__global__ void k_fc2(const float* __restrict__ z1,  // [BG,128]
                      const float* __restrict__ W,   // [1,128]
                      const float* __restrict__ bc,  // [1]
                      float* __restrict__ out) {     // [BG]
  int b = blockIdx.x * blockDim.x + threadIdx.x;
  if (b >= BG) return;
  float s = bc[0];
#pragma unroll 8
  for (int k = 0; k < 128; ++k) s += z1[b * 128 + k] * W[k];
  out[b] = s;
}

// ---------------- host driver ----------------
extern "C" void kernel_launch(void* const* d_in, const int* in_sizes, int n_in,
                              void* d_out, int out_size, void* d_ws, size_t ws_size,
                              hipStream_t stream) {
  (void)in_sizes; (void)n_in; (void)out_size; (void)ws_size;
  const float* x     = (const float*)d_in[0];
  const int*   eidx  = (const int*)d_in[1];
  const float* ea    = (const float*)d_in[2];
  const float* feats = (const float*)d_in[4];
  const int* esrc = eidx;
  const int* edst = eidx + NE;
  auto P = [&](int i) { return (const float*)d_in[i]; };

  float* ws = (float*)d_ws;
  const size_t NB = (size_t)NN * 256;
  float* h0    = ws;               // |--- these three are overlaid by GI ---|
  float* aggb  = ws + NB;
  float* mid   = ws + 2 * NB;
  float* h1    = ws + 3 * NB;      // survives: final GINE output lands here
  float* GI    = ws;               // [NN,768] overlays h0/agg/mid after GINE
  float* hs    = ws + 4 * NB;      // [BG,128,256]
  float* GH    = ws + 5 * NB;      // [BG,768]
  float* hst   = GH + (size_t)BG * 768;     // [BG,256]
  float* scale = hst + (size_t)BG * 256;    // [256]
  float* shift = scale + 256;               // [256]
  float* gpool = shift + 256;               // [BG,256]
  float* xd    = gpool + (size_t)BG * 256;  // [N7,256]
  float* xl    = xd + (size_t)N7 * 256;     // [N7,1024]
  float* xr    = xl + (size_t)N7 * 1024;    // [N7,1024]
  float* nout  = xr + (size_t)N7 * 1024;    // [N7,256]
  float* z1    = nout + (size_t)N7 * 256;   // [BG,128]

  auto gemm = [&](const float* A, const float* A2, const float* Wm, const float* bb,
                  float* Cm, int M, int Nn, int K, int lda, int act) {
    long waves = (long)(M / 16) * (Nn / 64);   // one wave per 16x64 block
    int grid = (int)((waves * 32 + 255) / 256);
    if (A2)
      k_wmma_gemm<true><<<grid, 256, 0, stream>>>(A, A2, Wm, bb, Cm, M, Nn, K, lda, act);
    else
      k_wmma_gemm<false><<<grid, 256, 0, stream>>>(A, nullptr, Wm, bb, Cm, M, Nn, K, lda, act);
  };
  auto zero = [&](float* p, long n) {
    k_zero<<<(int)((n + 255) / 256), 256, 0, stream>>>(p, n);
  };

  // ---- 4-layer GINE stack ----
  const float* hin = x;
  float* houts[4] = { h0, h1, h0, h1 };
  int dins[4] = { 64, 256, 256, 256 };
  for (int i = 0; i < 4; ++i) {
    int base = 5 + i * 8;  // ge_W, ge_b, W1, b1, W2, b2, bn_g, bn_b
    int din = dins[i];
    zero(aggb, (long)NN * din);
    long sth = (long)NE * (din / 8);
    k_gine_scatter<<<(int)((sth + 255) / 256), 256, 0, stream>>>(
        hin, esrc, edst, ea, P(base + 0), P(base + 1), aggb, NE, din);
    // mid = relu((h + agg) @ W1^T + b1)
    gemm(hin, aggb, P(base + 2), P(base + 3), mid, NN, 256, din, din, 1);
    // hout = mid @ W2^T + b2
    float* ho = houts[i];
    gemm(mid, nullptr, P(base + 4), P(base + 5), ho, NN, 256, 256, 256, 0);
    // hout = relu(BN(hout))
    k_bn_stats<<<256, 256, 0, stream>>>(ho, NN, 256, P(base + 6), P(base + 7), scale, shift);
    k_bn_relu<<<(int)((NB + 255) / 256), 256, 0, stream>>>(ho, (long)NB, 256, scale, shift);
    hin = ho;
  }
  // h_final == h1 (hin)

  // ---- GRU pooling ----
  gemm(hin, nullptr, P(37), P(38), GI, NN, 768, 256, 256, 0);   // GI = H @ Wih^T + bih
  zero(hst, (long)BG * 256);
  for (int t = 0; t < TT; ++t) {
    gemm(hst, nullptr, P(39), P(40), GH, BG, 768, 256, 256, 0); // GH = h @ Whh^T + bhh
    k_gru_gate<<<(BG * 256) / 256, 256, 0, stream>>>(GI, GH, hst, hs, t);
  }
  k_attn_pool<<<BG, 256, 0, stream>>>(hs, P(41), P(42), gpool);

  // ---- GATv2 over star graphs ----
  k_build_xd<<<(N7 * 256 + 255) / 256, 256, 0, stream>>>(gpool, feats, xd);
  gemm(xd, nullptr, P(43), P(44), xl, N7, 1024, 256, 256, 0);
  gemm(xd, nullptr, P(45), P(46), xr, N7, 1024, 256, 256, 0);
  k_gat_node<<<N7, 256, 0, stream>>>(xl, xr, P(47), P(48), nout);
  k_bn_stats<<<256, 256, 0, stream>>>(nout, N7, 256, P(49), P(50), scale, shift);
  k_bn_relu<<<(N7 * 256 + 255) / 256, 256, 0, stream>>>(nout, (long)N7 * 256, 256, scale, shift);

  // ---- head: central rows (stride 7*256) -> FC1 relu -> FC2 ----
  gemm(nout, nullptr, P(51), P(52), z1, BG, 128, 256, 7 * 256, 1);
  k_fc2<<<1, 256, 0, stream>>>(z1, P(53), P(54), (float*)d_out);
}